// Aligner_78408922956541
// MI455X (gfx1250) — compile-verified
//
#include <hip/hip_runtime.h>
#include <hip/hip_bf16.h>

// Problem constants (match reference)
#define B_   64
#define C_   192
#define T_Y_ 2000
#define T_X_ 400
#define KDIM (2*C_)          // 384: interleaved (o_p, -2*m_p*o_p) vs (z^2, z)
#define BITW 13              // ceil(400/32) decision-bit words per DP row
#define NEGV (-1e9f)
#define LOG2PI 1.8378770664093453f

// d_out layout (floats, concatenated in reference return order)
#define OUT_POOL 0                                  // [B, T_X, C]
#define OUT_DUR  ((size_t)B_*T_X_*C_)               // [B, T_X]
#define OUT_LOSS (OUT_DUR + (size_t)B_*T_X_)        // scalar
#define OUT_MASK (OUT_LOSS + 1)                     // [B, T_X]

typedef float v2f __attribute__((ext_vector_type(2)));
typedef float v8f __attribute__((ext_vector_type(8)));

// ---------------------------------------------------------------------------
// 1) Ragged lengths from masks. x_mask is float {0,1}; y_mask is numpy bool
//    (1 byte per element).
// ---------------------------------------------------------------------------
__global__ void lengths_kernel(const float* __restrict__ x_mask,
                               const unsigned char* __restrict__ y_mask,
                               int* __restrict__ txl, int* __restrict__ tyl) {
  const int b = blockIdx.x, tid = threadIdx.x;
  __shared__ int sx[256], sy[256];
  int cx = 0, cy = 0;
  for (int i = tid; i < T_X_; i += 256) cx += (x_mask[(size_t)b*T_X_ + i] != 0.0f);
  for (int i = tid; i < T_Y_; i += 256) cy += (y_mask[(size_t)b*T_Y_ + i] != 0);
  sx[tid] = cx; sy[tid] = cy;
  __syncthreads();
  for (int s = 128; s > 0; s >>= 1) {
    if (tid < s) { sx[tid] += sx[tid + s]; sy[tid] += sy[tid + s]; }
    __syncthreads();
  }
  if (tid == 0) { txl[b] = sx[0]; tyl[b] = sy[0]; }
}

// ---------------------------------------------------------------------------
// 2) Pack A[b,x,k] = interleave(o_p, -2*m_p*o_p) along k=2c{,+1}, and
//    bias[b,x] = sum_c (m_p^2*o_p + 2*logs_p)   (= term_c + term_d)
// ---------------------------------------------------------------------------
__global__ void apack_kernel(const float* __restrict__ m_p,
                             const float* __restrict__ logs_p,
                             float* __restrict__ Apack, float* __restrict__ bias) {
  const int t = blockIdx.x * blockDim.x + threadIdx.x;
  if (t >= B_ * T_X_) return;
  const int b = t / T_X_, x = t % T_X_;
  const float* mp = m_p    + (size_t)b*C_*T_X_ + x;
  const float* lp = logs_p + (size_t)b*C_*T_X_ + x;
  float* ap = Apack + ((size_t)b*T_X_ + x) * KDIM;
  float bacc = 0.f;
  for (int c = 0; c < C_; ++c) {
    const float ls = lp[(size_t)c*T_X_];
    const float o  = expf(-2.f*ls);          // precision, o_p
    const float m  = mp[(size_t)c*T_X_];
    ap[2*c]   = o;
    ap[2*c+1] = -2.f*m*o;
    bacc += m*m*o + 2.f*ls;
  }
  bias[(size_t)b*T_X_ + x] = bacc;
}

// ---------------------------------------------------------------------------
// 3) WMMA GEMM: log_p[b,y,x] = (x<txl) ? -0.5*(sum_k A[x,k]*Bv[k,y]+bias[x])
//                                       : NEG
//    Bv[2c,y]=z^2, Bv[2c+1,y]=z built in-register from z_flow.
//    One wave -> 16x64 strip via V_WMMA_F32_16X16X4_F32, 96 K-steps.
//    f32 16x16x4 layout: A lane(L): M=L&15, VGPR0 holds K={k0,k0+2},
//    VGPR1 K={k0+1,k0+3} (halves by L>>4); B mirrors with N=L&15.
//    With k0=4t, a lane's two K values share channel c=2t+(L>>4), so the
//    B fragment is simply {z*z, z} from a single z load.
//    The K-loop is branch-free: out-of-range y-tiles clamp their load offset
//    to tile 0 (valid address) and compute garbage that is never stored.
// ---------------------------------------------------------------------------
__global__ __launch_bounds__(128)
void logp_gemm(const float* __restrict__ Apack, const float* __restrict__ bias,
               const float* __restrict__ z_flow, const int* __restrict__ txl,
               float* __restrict__ logp) {
  const int b    = blockIdx.y;
  const int lane = threadIdx.x & 31;
  const int wid  = threadIdx.x >> 5;
  const int NSTRIP = (T_X_/16) * ((T_Y_ + 63) / 64);   // 25 * 32 = 800
  const int strip = blockIdx.x * 4 + wid;
  if (strip >= NSTRIP) return;
  const int xt = strip % (T_X_/16);
  const int ys = strip / (T_X_/16);
  const int x_base = xt * 16, y_base = ys * 64;
  const int n = lane & 15, half = lane >> 4;
  const int jmax = min(4, (T_Y_ - y_base) >> 4);

  // Clamped per-tile y offsets: keep every load in range, no branches in loop.
  int yo[4];
#pragma unroll
  for (int j = 0; j < 4; ++j) yo[j] = (j < jmax) ? 16*j : 0;

  v8f acc[4] = {};
  const float* ap = Apack + ((size_t)b*T_X_ + x_base + n) * KDIM + 2*half;
  const float* zp = z_flow + (size_t)b*C_*T_Y_ + (y_base + n);

#pragma unroll 2
  for (int t = 0; t < C_/2; ++t) {                 // K-step: K0 = 4t
    const v2f a = *(const v2f*)(ap + 4*t);
    const float* zrow = zp + (size_t)(2*t + half) * T_Y_;
#pragma unroll
    for (int j = 0; j < 4; ++j) {
      const float z = zrow[yo[j]];
      v2f bf; bf.x = z*z; bf.y = z;
      acc[j] = __builtin_amdgcn_wmma_f32_16x16x4_f32(
          false, a, false, bf, (short)0, acc[j], false, false);
    }
  }

  const int txl_b = txl[b];
  const float4 bq0 = *(const float4*)(bias + (size_t)b*T_X_ + x_base + 8*half);
  const float4 bq1 = *(const float4*)(bias + (size_t)b*T_X_ + x_base + 8*half + 4);
  const float bv[8] = {bq0.x, bq0.y, bq0.z, bq0.w, bq1.x, bq1.y, bq1.z, bq1.w};

#pragma unroll
  for (int j = 0; j < 4; ++j) {
    if (j < jmax) {
      const int y = y_base + 16*j + n;
      float* op = logp + ((size_t)b*T_Y_ + y) * T_X_ + x_base + 8*half;
      float vals[8];
#pragma unroll
      for (int r = 0; r < 8; ++r) {
        const int x = x_base + 8*half + r;
        vals[r] = (x < txl_b) ? -0.5f*(acc[j][r] + bv[r]) : NEGV;
      }
      *(float4*)(op)     = make_float4(vals[0], vals[1], vals[2], vals[3]);
      *(float4*)(op + 4) = make_float4(vals[4], vals[5], vals[6], vals[7]);
    }
  }
}

// ---------------------------------------------------------------------------
// 4) MAS: forward Viterbi DP over 2000 rows (row in LDS, 1 thread per x),
//    decision bits via wave32 ballot, then scalar backtrack + duration
//    prefix by thread 0.
// ---------------------------------------------------------------------------
__global__ __launch_bounds__(416)
void mas_kernel(const float* __restrict__ logp, const int* __restrict__ txl,
                const int* __restrict__ tyl, unsigned int* __restrict__ bits,
                int* __restrict__ path, int* __restrict__ dur,
                int* __restrict__ segstart) {
  const int b = blockIdx.x;
  const int x = threadIdx.x;
  const bool active = x < T_X_;
  __shared__ float v[T_X_];
  __shared__ int sdur[T_X_];
  const int txl_b = txl[b], tyl_b = tyl[b];
  if (active) { v[x] = (x == 0) ? 0.f : NEGV; sdur[x] = 0; }
  __syncthreads();

  const float* lrow = logp + (size_t)b*T_Y_*T_X_ + x;
  unsigned int* brow = bits + (size_t)b*T_Y_*BITW;

  for (int y = 0; y < T_Y_; ++y) {
    float vc = 0.f, vl = NEGV, lp = 0.f;
    if (active) {
      vc = v[x];
      vl = (x > 0) ? v[x-1] : NEGV;
      lp = lrow[(size_t)y*T_X_];
      if (y + 16 < T_Y_) __builtin_prefetch(lrow + (size_t)(y+16)*T_X_, 0, 1);
    }
    // bit[y][x] = value[y-1][x] < value[y-1][x-1]  (backtrack decision)
    const unsigned long long bal = __ballot(active && (vc < vl));
    if ((x & 31) == 0) {
      const int w = x >> 5;
      if (w < BITW) brow[(size_t)y*BITW + w] = (unsigned int)bal;
    }
    __syncthreads();
    if (active) {
      const float vn = lp + fmaxf(vc, vl);
      v[x] = ((x <= y) && (x < txl_b)) ? vn : NEGV;
    }
    __syncthreads();
  }

  if (threadIdx.x == 0) {
    int idx = txl_b - 1;
    for (int y = T_Y_ - 1; y >= 0; --y) {
      if (y < tyl_b) {
        path[(size_t)b*T_Y_ + y] = idx;
        sdur[idx] += 1;
        const bool dec = (idx != 0) &&
            ((idx == y) ||
             ((brow[(size_t)y*BITW + (idx >> 5)] >> (idx & 31)) & 1u));
        if (dec) --idx;
      } else {
        path[(size_t)b*T_Y_ + y] = -1;
      }
    }
    int off = 0;
    for (int xx = 0; xx < T_X_; ++xx) { segstart[b*T_X_ + xx] = off; off += sdur[xx]; }
  }
  __syncthreads();
  if (active) dur[b*T_X_ + x] = sdur[x];
}

// ---------------------------------------------------------------------------
// 5) Segment-mean pooling + durations + text_mask outputs.
// ---------------------------------------------------------------------------
__global__ void pool_kernel(const float* __restrict__ z_spec,
                            const int* __restrict__ txl,
                            const int* __restrict__ dur,
                            const int* __restrict__ segstart,
                            float* __restrict__ out) {
  const int x = blockIdx.x, b = blockIdx.y, c = threadIdx.x;   // blockDim = C_
  const int d = dur[b*T_X_ + x];
  const int s = segstart[b*T_X_ + x];
  const float* zs = z_spec + ((size_t)b*C_ + c) * T_Y_;
  float sum = 0.f;
  for (int i = 0; i < d; ++i) sum += zs[s + i];
  out[OUT_POOL + ((size_t)b*T_X_ + x)*C_ + c] = sum / (float)max(d, 1);
  if (c == 0) {
    out[OUT_DUR  + (size_t)b*T_X_ + x] = (float)d;
    out[OUT_MASK + (size_t)b*T_X_ + x] = (x < txl[b]) ? 0.0f : 1.0f;
  }
}

// ---------------------------------------------------------------------------
// 6) NLL loss: deterministic two-stage reduction (no float atomics).
// ---------------------------------------------------------------------------
#define YCHUNK 250
__global__ void loss_partial(const float* __restrict__ z_flow,
                             const float* __restrict__ m_p,
                             const float* __restrict__ logs_p,
                             const int* __restrict__ tyl,
                             const int* __restrict__ path,
                             float* __restrict__ partials) {
  const int b = blockIdx.x, ch = blockIdx.y, tid = threadIdx.x;
  const int tyl_b = tyl[b];
  __shared__ float red[256];
  float local = 0.f;
  for (int idx = tid; idx < C_*YCHUNK; idx += 256) {
    const int yloc = idx % YCHUNK, c = idx / YCHUNK;
    const int y = ch*YCHUNK + yloc;
    if (y < tyl_b) {
      const int px = path[(size_t)b*T_Y_ + y];
      const float m  = m_p   [((size_t)b*C_ + c)*T_X_ + px];
      const float ls = logs_p[((size_t)b*C_ + c)*T_X_ + px];
      const float iv = expf(-2.f*ls);
      const float dz = z_flow[((size_t)b*C_ + c)*T_Y_ + y] - m;
      local += 0.5f*(dz*dz*iv + 2.f*ls + LOG2PI);
    }
  }
  red[tid] = local;
  __syncthreads();
  for (int s = 128; s > 0; s >>= 1) {
    if (tid < s) red[tid] += red[tid + s];
    __syncthreads();
  }
  if (tid == 0) partials[b*8 + ch] = red[0];
}

__global__ void loss_final(const float* __restrict__ partials,
                           const float* __restrict__ logdet,
                           const int* __restrict__ tyl,
                           float* __restrict__ out) {
  __shared__ float red[512];
  const int tid = threadIdx.x;
  red[tid] = partials[tid];                    // exactly B_*8 = 512
  __syncthreads();
  for (int s = 256; s > 0; s >>= 1) {
    if (tid < s) red[tid] += red[tid + s];
    __syncthreads();
  }
  if (tid == 0) {
    float ld = 0.f; long ty = 0;
    for (int b = 0; b < B_; ++b) { ld += logdet[b]; ty += tyl[b]; }
    const float denom = fmaxf((float)ty * (float)C_, 1.0f);
    out[OUT_LOSS] = (red[0] - ld) / denom;
  }
}

// ---------------------------------------------------------------------------
// Host launcher
// ---------------------------------------------------------------------------
extern "C" void kernel_launch(void* const* d_in, const int* in_sizes, int n_in,
                              void* d_out, int out_size, void* d_ws, size_t ws_size,
                              hipStream_t stream) {
  (void)in_sizes; (void)n_in; (void)out_size; (void)ws_size;
  const float* z_spec = (const float*)d_in[0];
  const float* z_flow = (const float*)d_in[1];
  const float* logdet = (const float*)d_in[2];
  const float* m_p    = (const float*)d_in[3];
  const float* logs_p = (const float*)d_in[4];
  const float* x_mask = (const float*)d_in[5];
  const unsigned char* y_mask = (const unsigned char*)d_in[6];   // numpy bool
  float* out = (float*)d_out;

  char* ws = (char*)d_ws;
  size_t off = 0;
  auto take = [&](size_t bytes) -> char* {
    char* p = ws + off;
    off = (off + bytes + 255) & ~(size_t)255;
    return p;
  };
  int*   txl      = (int*)  take((size_t)B_*sizeof(int));
  int*   tyl      = (int*)  take((size_t)B_*sizeof(int));
  float* Apack    = (float*)take((size_t)B_*T_X_*KDIM*sizeof(float));     // 39.3 MB
  float* bias     = (float*)take((size_t)B_*T_X_*sizeof(float));
  float* logp     = (float*)take((size_t)B_*T_Y_*T_X_*sizeof(float));     // 204.8 MB
  unsigned int* bits = (unsigned int*)take((size_t)B_*T_Y_*BITW*sizeof(unsigned int)); // 6.7 MB
  int*   path     = (int*)  take((size_t)B_*T_Y_*sizeof(int));
  int*   dur      = (int*)  take((size_t)B_*T_X_*sizeof(int));
  int*   segstart = (int*)  take((size_t)B_*T_X_*sizeof(int));
  float* partials = (float*)take((size_t)512*sizeof(float));

  lengths_kernel<<<B_, 256, 0, stream>>>(x_mask, y_mask, txl, tyl);
  apack_kernel<<<(B_*T_X_ + 255)/256, 256, 0, stream>>>(m_p, logs_p, Apack, bias);
  logp_gemm<<<dim3(200, B_), 128, 0, stream>>>(Apack, bias, z_flow, txl, logp);
  mas_kernel<<<B_, 416, 0, stream>>>(logp, txl, tyl, bits, path, dur, segstart);
  pool_kernel<<<dim3(T_X_, B_), C_, 0, stream>>>(z_spec, txl, dur, segstart, out);
  loss_partial<<<dim3(B_, 8), 256, 0, stream>>>(z_flow, m_p, logs_p, tyl, path, partials);
  loss_final<<<1, 512, 0, stream>>>(partials, logdet, tyl, out);
}